// GINSampling_18141941859014
// MI455X (gfx1250) — compile-verified
//
#include <hip/hip_runtime.h>
#include <hip/hip_bf16.h>

typedef __attribute__((ext_vector_type(2))) float v2f;
typedef __attribute__((ext_vector_type(8))) float v8f;

#define GIN_N_NODES 50000
#define GIN_N_EDGES 600000
#define GIN_D_FEAT  128
#define GIN_BN_EPS  1e-5f

// ---------------- degree counts: cnt[dst] += 1 ----------------
__global__ void gin_count_kernel(const int* __restrict__ dst, float* __restrict__ cnt, int n_edges) {
    int e = blockIdx.x * blockDim.x + threadIdx.x;
    if (e < n_edges) atomicAdd(&cnt[dst[e]], 1.0f);
}

// ---------------- per-edge gather + scatter-add, one wave32 per edge, D=128 ----------------
__global__ void gin_aggregate_kernel(const float* __restrict__ h, const int* __restrict__ src,
                                     const int* __restrict__ dst, float* __restrict__ acc,
                                     int n_edges) {
    int wid  = (blockIdx.x * blockDim.x + threadIdx.x) >> 5;   // wave id == edge id
    int lane = threadIdx.x & 31;
    if (wid >= n_edges) return;
    int s = __builtin_amdgcn_readfirstlane(src[wid]);
    int t = __builtin_amdgcn_readfirstlane(dst[wid]);
    const float4 v = ((const float4*)(h + (size_t)s * GIN_D_FEAT))[lane];  // 32 lanes x 16B = 512B
    float* o = acc + (size_t)t * GIN_D_FEAT + lane * 4;
    atomicAdd(o + 0, v.x);
    atomicAdd(o + 1, v.y);
    atomicAdd(o + 2, v.z);
    atomicAdd(o + 3, v.w);
}

// ---------------- x = h + sum * (1/max(cnt,1)), in-place on sum buffer ----------------
__global__ void gin_combine_kernel(const float* __restrict__ h, float* __restrict__ x,
                                   const float* __restrict__ cnt, int total) {
    int i = blockIdx.x * blockDim.x + threadIdx.x;
    if (i >= total) return;
    int row = i >> 7;  // / 128
    float inv = 1.0f / fmaxf(cnt[row], 1.0f);
    x[i] = h[i] + x[i] * inv;
}

// ---------------- Y = X @ W + b  via V_WMMA_F32_16X16X4_F32, plus BN col-stats ----------------
// grid.x = n_rows/16 ; blockDim.x = 32 * (dout/16)  (256 for dout=128, 128 for dout=64)
__global__ __launch_bounds__(256) void gin_gemm_wmma_kernel(
    const float* __restrict__ X, const float* __restrict__ W, const float* __restrict__ bias,
    float* __restrict__ Y, float* __restrict__ stats, int din, int dout) {
    __shared__ float sA[16 * 129];      // 16-row A strip, padded stride (bank-conflict free)
    __shared__ float sWt[128 * 129];    // W stored COLUMN-major: sWt[n*(din+1)+k]
    __shared__ float sSum[128];
    __shared__ float sSq[128];

    const int tid  = threadIdx.x;
    const int nthr = blockDim.x;
    const int r0   = blockIdx.x * 16;
    const int lda  = din + 1;   // padded stride for both sA rows and sWt columns
    const int ldw  = din + 1;

    for (int i = tid; i < dout; i += nthr) { sSum[i] = 0.0f; sSq[i] = 0.0f; }
    // transpose W (din x dout, row-major) into sWt: coalesced global reads,
    // LDS writes stride ldw -> bank step (129 mod 64)=1, conflict-free
    for (int i = tid; i < din * dout; i += nthr) {
        int k = i / dout, n = i - k * dout;
        sWt[n * ldw + k] = W[i];
    }
    for (int i = tid; i < 16 * din; i += nthr) {
        int m = i / din, k = i - m * din;
        sA[m * lda + k] = X[(size_t)(r0 + m) * din + k];
    }
    __syncthreads();

    const int wave = tid >> 5;
    const int lane = tid & 31;
    const int half = lane >> 4;      // A/B: K-pair select ; C: +8 row offset
    const int mn   = lane & 15;      // A: row M ; B/C: col N
    const int n0   = wave * 16;

    // loop-invariant lane bases -> all K-loop DS loads become immediate-offset 2addr loads
    const float* aBase = &sA[mn * lda];
    const float* bBase = &sWt[(n0 + mn) * ldw];

    v8f acc = {};
    for (int k0 = 0; k0 < din; k0 += 4) {
        const int kk = k0 + 2 * half;
        v2f a = {}, b = {};
        a.x = aBase[kk];
        a.y = aBase[kk + 1];
        b.x = bBase[kk];
        b.y = bBase[kk + 1];
        // D = A(16x4,f32) * B(4x16,f32) + C(16x16,f32)
        acc = __builtin_amdgcn_wmma_f32_16x16x4_f32(false, a, false, b, (short)0, acc, false, false);
    }

    const float bv = bias[n0 + mn];
    float* yrow = Y + (size_t)r0 * dout;
#pragma unroll
    for (int v = 0; v < 8; ++v) {
        const int m = v + 8 * half;                     // C layout: VGPR v -> rows v / v+8
        const float val = acc[v] + bv;
        yrow[(size_t)m * dout + n0 + mn] = val;
        atomicAdd(&sSum[n0 + mn], val);                 // ds_add_f32 block-level reduction
        atomicAdd(&sSq[n0 + mn], val * val);
    }
    __syncthreads();
    for (int i = tid; i < dout; i += nthr) {
        atomicAdd(&stats[i], sSum[i]);
        atomicAdd(&stats[dout + i], sSq[i]);
    }
}

// ---------------- y = relu(gamma*(y-mu)*rsqrt(var+eps)+beta), in-place ----------------
__global__ void gin_bn_relu_kernel(float* __restrict__ Y, const float* __restrict__ stats,
                                   const float* __restrict__ gamma, const float* __restrict__ beta,
                                   int total, int dout) {
    int i = blockIdx.x * blockDim.x + threadIdx.x;
    if (i >= total) return;
    int c = i & (dout - 1);  // dout is 128 or 64
    const float invn = 1.0f / (float)GIN_N_NODES;
    float mu  = stats[c] * invn;
    float var = stats[dout + c] * invn - mu * mu;   // biased variance, as reference
    float scale = gamma[c] * rsqrtf(var + GIN_BN_EPS);
    float v = (Y[i] - mu) * scale + beta[c];
    Y[i] = fmaxf(v, 0.0f);
}

extern "C" void kernel_launch(void* const* d_in, const int* in_sizes, int n_in,
                              void* d_out, int out_size, void* d_ws, size_t ws_size,
                              hipStream_t stream) {
    const float* features = (const float*)d_in[0];
    const int*   src      = (const int*)d_in[1];
    const int*   dst      = (const int*)d_in[2];

    const size_t NB = (size_t)GIN_N_NODES * GIN_D_FEAT;  // 6.4M floats
    float* buf0  = (float*)d_ws;          // neigh-sum / x buffer
    float* buf1  = buf0 + NB;             // y buffer
    float* buf2  = buf1 + NB;             // layer output (next h)
    float* cnt   = buf2 + NB;             // [N] degree counts
    float* stats = cnt + GIN_N_NODES;     // [2*128] col sums / sumsq

    const float* h = features;
    for (int li = 0; li < 3; ++li) {
        const float* W1  = (const float*)d_in[3 + 8 * li + 0];
        const float* b1  = (const float*)d_in[3 + 8 * li + 1];
        const float* g1  = (const float*)d_in[3 + 8 * li + 2];
        const float* be1 = (const float*)d_in[3 + 8 * li + 3];
        const float* W2  = (const float*)d_in[3 + 8 * li + 4];
        const float* b2  = (const float*)d_in[3 + 8 * li + 5];
        const float* g2  = (const float*)d_in[3 + 8 * li + 6];
        const float* be2 = (const float*)d_in[3 + 8 * li + 7];
        const int d1 = (li == 2) ? 64 : 128;   // W1: 128 -> d1 ; W2: d1 -> d1

        // ---- mean aggregation: x = h + segment_mean(h[src], dst) ----
        hipMemsetAsync(buf0, 0, NB * sizeof(float), stream);
        hipMemsetAsync(cnt, 0, GIN_N_NODES * sizeof(float), stream);
        gin_count_kernel<<<(GIN_N_EDGES + 255) / 256, 256, 0, stream>>>(dst, cnt, GIN_N_EDGES);
        gin_aggregate_kernel<<<(GIN_N_EDGES * 32) / 256, 256, 0, stream>>>(h, src, dst, buf0, GIN_N_EDGES);
        gin_combine_kernel<<<((int)NB + 255) / 256, 256, 0, stream>>>(h, buf0, cnt, (int)NB);

        // ---- MLP layer 1: relu(bn(x @ W1 + b1)) ----
        hipMemsetAsync(stats, 0, 256 * sizeof(float), stream);
        gin_gemm_wmma_kernel<<<GIN_N_NODES / 16, 32 * (d1 / 16), 0, stream>>>(
            buf0, W1, b1, buf1, stats, 128, d1);
        gin_bn_relu_kernel<<<(GIN_N_NODES * d1 + 255) / 256, 256, 0, stream>>>(
            buf1, stats, g1, be1, GIN_N_NODES * d1, d1);

        // ---- MLP layer 2: relu(bn(y @ W2 + b2)) ----
        float* out2 = (li == 2) ? (float*)d_out : buf2;
        hipMemsetAsync(stats, 0, 256 * sizeof(float), stream);
        gin_gemm_wmma_kernel<<<GIN_N_NODES / 16, 32 * (d1 / 16), 0, stream>>>(
            buf1, W2, b2, out2, stats, d1, d1);
        gin_bn_relu_kernel<<<(GIN_N_NODES * d1 + 255) / 256, 256, 0, stream>>>(
            out2, stats, g2, be2, GIN_N_NODES * d1, d1);
        h = out2;
    }
}